// Custom_Integration_layer_79817672228971
// MI455X (gfx1250) — compile-verified
//
#include <hip/hip_runtime.h>
#include <hip/hip_bf16.h>

// Problem constants from the reference: B=2048, P=4, H=16, K=48, D=4.
#define BDIM 2048
#define PDIM 4
#define BP   (BDIM * PDIM)   // 8192 independent problems
#define HV   16
#define KV   48
#define GV   64              // H + K Gaussians
#define DV   4
#define MS   17              // padded LDS row stride for 16x16 matrices (bank-conflict-free columns)
#define WPB  8               // waves per block (wave32) -> 256 threads
#define LOG2PI_F 1.8378770664093453f

typedef __attribute__((ext_vector_type(2))) float v2f;
typedef __attribute__((ext_vector_type(8))) float v8f;

struct __align__(16) WaveScratch {
  float c[GV];          // stacked biases
  float M1[HV * MS];    // A^T A  -> Cholesky L1 (in place)
  float GAB[HV * MS];   // A^T B
  float X[HV * MS];     // L1^{-1} (A^T B)
  float S[HV * MS];     // B^T B - X^T X -> Cholesky L2 (in place); R2 = L2^T
  float vAc[HV];        // A^T c
  float w[HV];          // B^T c, then  B^T c - X^T y
  float y[HV];          // L1^{-1} (A^T c)
  float nb[HV];         // L2^{-1} w  (kept biases)
};

// D = A*B + C on the matrix pipe:  v_wmma_f32_16x16x4_f32
static __device__ __forceinline__ v8f wmma_f32x4(v2f a, v2f b, v8f c) {
  return __builtin_amdgcn_wmma_f32_16x16x4_f32(false, a, false, b, (short)0, c,
                                               false, false);
}

// Scatter a 16x16 f32 C/D accumulator (8 VGPRs) to LDS.
// C/D layout: VGPR i -> (M=i, N=lane) for lanes 0-15, (M=8+i, N=lane-16) for lanes 16-31.
static __device__ __forceinline__ void store_cd16(float* m, v8f acc, int lane) {
  const int colN = lane & 15;
  const int mb   = (lane < 16) ? 0 : 8;
#pragma unroll
  for (int i = 0; i < 8; ++i) m[(mb + i) * MS + colN] = acc[i];
}

// In-place lane-parallel Cholesky of a 16x16 SPD matrix held in LDS (lower triangle -> L).
// Same-wave LDS ops are in-order (DS queue); wave_barrier is a compiler fence.
static __device__ void chol16(float* M, int lane) {
#pragma unroll 1
  for (int k = 0; k < HV; ++k) {
    __builtin_amdgcn_wave_barrier();
    const float dk  = M[k * MS + k];
    const float lkk = sqrtf(dk);
    const float inv = 1.0f / lkk;
    if (lane == k) M[k * MS + k] = lkk;
    float ljk = 0.0f;
    const bool act = (lane > k) && (lane < HV);
    if (act) { ljk = M[lane * MS + k] * inv; M[lane * MS + k] = ljk; }
    __builtin_amdgcn_wave_barrier();
    if (act) {
      for (int i = k + 1; i <= lane; ++i)
        M[lane * MS + i] = fmaf(-ljk, M[i * MS + k], M[lane * MS + i]);
    }
  }
}

__global__ __launch_bounds__(WPB * 32)
void integ_kernel(const float* __restrict__ input_i,      // [B,P,D]
                  const float* __restrict__ prev_coefs,   // [H,B,P,H]
                  const float* __restrict__ prev_biases,  // [H,B,P]
                  const float* __restrict__ lp,           // [B,P]
                  const float* __restrict__ log_factors,  // [B,P]
                  const float* __restrict__ obs_coefs,    // [K,B,P,D]
                  const float* __restrict__ hid_coefs,    // [K,B,P,H]
                  const float* __restrict__ nxt_coefs,    // [K,B,P,H]
                  const float* __restrict__ rec_biases,   // [K,B,P]
                  float* __restrict__ out_coefs,          // [H,B,P,H]
                  float* __restrict__ out_biases,         // [H,B,P]
                  float* __restrict__ out_lp)             // [B,P]
{
  __shared__ WaveScratch smem[WPB];
  const int lane = threadIdx.x & 31;
  const int wid  = threadIdx.x >> 5;
  const int bp   = blockIdx.x * WPB + wid;   // grid sized exactly: no tail
  WaveScratch* W = &smem[wid];

  // ---------------- stacked biases c[0..63] -------------------------------
  const float4 xin = *(const float4*)(input_i + (size_t)bp * DV);
  if (lane < HV) W->c[lane] = prev_biases[(size_t)lane * BP + bp];
  for (int k = lane; k < KV; k += 32) {
    const float4 o = *(const float4*)(obs_coefs + ((size_t)k * BP + bp) * DV);
    W->c[HV + k] = rec_biases[(size_t)k * BP + bp]
                 + o.x * xin.x + o.y * xin.y + o.z * xin.z + o.w * xin.w;
  }
  __builtin_amdgcn_wave_barrier();

  // cc = c^T c (full-wave butterfly)
  float cc = W->c[lane] * W->c[lane] + W->c[lane + 32] * W->c[lane + 32];
#pragma unroll
  for (int off = 16; off >= 1; off >>= 1) cc += __shfl_xor(cc, off, 32);

  // ---------------- Gram products on the matrix pipe ----------------------
  // Per K-chunk of 4 rows: lane L<16 holds rows g0+0,g0+1 at col L; lanes 16-31
  // hold rows g0+2,g0+3 at col L-16 — this register pattern is simultaneously a
  // valid 16x4 A-operand and 4x16 B-operand for v_wmma_f32_16x16x4_f32.
  const int    col   = lane & 15;
  const int    rsel  = (lane < 16) ? 0 : 2;
  const size_t SBH   = (size_t)BP * HV;           // stride between Gaussian rows
  const size_t cbase = (size_t)bp * HV + col;
  v8f m1 = {}, gab = {}, s0 = {};
  float pAc = 0.0f, pBc = 0.0f;

#pragma unroll
  for (int ci = 0; ci < 4; ++ci) {                // rows 0..15: prev Gaussians (Bm rows are 0)
    const int r0 = 4 * ci + rsel;
    v2f a;
    a.x = prev_coefs[(size_t)r0 * SBH + cbase];
    a.y = prev_coefs[(size_t)(r0 + 1) * SBH + cbase];
    const float2 cp = *(const float2*)&W->c[r0]; // r0 even -> 8B aligned
    pAc = fmaf(a.x, cp.x, fmaf(a.y, cp.y, pAc));
    m1 = wmma_f32x4(a, a, m1);
  }
#pragma unroll
  for (int ci = 0; ci < 12; ++ci) {               // rows 16..63: recurrent Gaussians
    const int r0 = 4 * ci + rsel;
    v2f a, b;
    a.x = hid_coefs[(size_t)r0 * SBH + cbase];
    a.y = hid_coefs[(size_t)(r0 + 1) * SBH + cbase];
    b.x = nxt_coefs[(size_t)r0 * SBH + cbase];
    b.y = nxt_coefs[(size_t)(r0 + 1) * SBH + cbase];
    const float2 cp = *(const float2*)&W->c[HV + r0];
    pAc = fmaf(a.x, cp.x, fmaf(a.y, cp.y, pAc));
    pBc = fmaf(b.x, cp.x, fmaf(b.y, cp.y, pBc));
    m1  = wmma_f32x4(a, a, m1);                   // A^T A
    gab = wmma_f32x4(a, b, gab);                  // A^T B
    s0  = wmma_f32x4(b, b, s0);                   // B^T B
  }

  // fold column halves of the matrix-vector partials
  pAc += __shfl_xor(pAc, 16, 32);
  pBc += __shfl_xor(pBc, 16, 32);
  if (lane < HV) { W->vAc[lane] = pAc; W->w[lane] = pBc; }

  store_cd16(W->M1, m1, lane);
  store_cd16(W->GAB, gab, lane);
  __builtin_amdgcn_wave_barrier();

  // ---------------- phase 1: Cholesky + projection ------------------------
  chol16(W->M1, lane);                            // M1 -> L1 ; LC1 = -sum log L1_ii
  __builtin_amdgcn_wave_barrier();

  // forward solve L1 * X = GAB (lane n<16 owns column n); lane 16 solves L1*y = vAc
  float xcol[HV];
#pragma unroll
  for (int r = 0; r < HV; ++r) {
    float s = (lane < HV) ? W->GAB[r * MS + lane]
                          : ((lane == HV) ? W->vAc[r] : 0.0f);
#pragma unroll
    for (int t = 0; t < r; ++t) s = fmaf(-W->M1[r * MS + t], xcol[t], s);
    xcol[r] = s / W->M1[r * MS + r];
  }
  if (lane < HV) {
#pragma unroll
    for (int r = 0; r < HV; ++r) W->X[r * MS + lane] = xcol[r];
  } else if (lane == HV) {
#pragma unroll
    for (int r = 0; r < HV; ++r) W->y[r] = xcol[r];
  }
  __builtin_amdgcn_wave_barrier();

  // w = B^T c - X^T y   (xcol still holds column `lane` of X)
  if (lane < HV) {
    float acc = W->w[lane];
#pragma unroll
    for (int r = 0; r < HV; ++r) acc = fmaf(-xcol[r], W->y[r], acc);
    W->w[lane] = acc;
  }
  __builtin_amdgcn_wave_barrier();

  // ---------------- phase 2: S = B^T B - X^T X (X^T X via WMMA) -----------
  v8f xtx = {};
#pragma unroll
  for (int r0 = 0; r0 < HV; r0 += 4) {
    v2f xa;
    xa.x = W->X[(r0 + rsel) * MS + col];
    xa.y = W->X[(r0 + rsel + 1) * MS + col];
    xtx = wmma_f32x4(xa, xa, xtx);
  }
  v8f sreg;
#pragma unroll
  for (int i = 0; i < 8; ++i) sreg[i] = s0[i] - xtx[i];
  store_cd16(W->S, sreg, lane);
  __builtin_amdgcn_wave_barrier();

  chol16(W->S, lane);                             // S -> L2 ; R2 = L2^T
  __builtin_amdgcn_wave_barrier();

  // nb = L2^{-1} w  (tiny single-column solve; lane 0)
  if (lane == 0) {
#pragma unroll 1
    for (int r = 0; r < HV; ++r) {
      float s = W->w[r];
      for (int t = 0; t < r; ++t) s = fmaf(-W->S[r * MS + t], W->nb[t], s);
      W->nb[r] = s / W->S[r * MS + r];
    }
  }
  __builtin_amdgcn_wave_barrier();

  // ---------------- log-constants + outputs -------------------------------
  float yty = 0.0f, nn = 0.0f, lc = 0.0f;
#pragma unroll
  for (int i = 0; i < HV; ++i) {
    yty = fmaf(W->y[i],  W->y[i],  yty);
    nn  = fmaf(W->nb[i], W->nb[i], nn);
    lc -= __logf(W->M1[i * MS + i]);
  }
  const float resid = fmaxf(cc - yty - nn, 0.0f);   // ||Pc||^2 - ||nb||^2
  lc -= 0.5f * ((float)(KV - HV) * LOG2PI_F + resid);

  if (lane == 0) out_lp[bp] = lp[bp] + lc + log_factors[bp];
  if (lane < HV) out_biases[(size_t)lane * BP + bp] = W->nb[lane];

  // Next_coefs[i, bp, j] = R2[i][j] = (j >= i) ? L2[j][i] : 0 ; 2 float4 stores/lane
#pragma unroll
  for (int s2 = 0; s2 < 2; ++s2) {
    const int q  = lane * 2 + s2;   // 0..63
    const int i  = q >> 2;          // output row
    const int j0 = (q & 3) * 4;     // output col start
    float4 v;
    v.x = (j0 + 0 >= i) ? W->S[(j0 + 0) * MS + i] : 0.0f;
    v.y = (j0 + 1 >= i) ? W->S[(j0 + 1) * MS + i] : 0.0f;
    v.z = (j0 + 2 >= i) ? W->S[(j0 + 2) * MS + i] : 0.0f;
    v.w = (j0 + 3 >= i) ? W->S[(j0 + 3) * MS + i] : 0.0f;
    *(float4*)(out_coefs + (size_t)i * SBH + (size_t)bp * HV + j0) = v;
  }
}

extern "C" void kernel_launch(void* const* d_in, const int* in_sizes, int n_in,
                              void* d_out, int out_size, void* d_ws, size_t ws_size,
                              hipStream_t stream) {
  (void)in_sizes; (void)n_in; (void)out_size; (void)d_ws; (void)ws_size;
  const float* input_i     = (const float*)d_in[0];
  const float* prev_coefs  = (const float*)d_in[1];
  const float* prev_biases = (const float*)d_in[2];
  const float* lp          = (const float*)d_in[3];
  const float* log_factors = (const float*)d_in[4];
  const float* obs_coefs   = (const float*)d_in[5];
  const float* hid_coefs   = (const float*)d_in[6];
  const float* nxt_coefs   = (const float*)d_in[7];
  const float* rec_biases  = (const float*)d_in[8];

  float* out_coefs  = (float*)d_out;                          // [H,B,P,H]
  float* out_biases = out_coefs + (size_t)HV * BP * HV;       // [H,B,P]
  float* out_lp     = out_biases + (size_t)HV * BP;           // [B,P]

  dim3 grid(BP / WPB);
  dim3 block(WPB * 32);
  integ_kernel<<<grid, block, 0, stream>>>(input_i, prev_coefs, prev_biases, lp,
                                           log_factors, obs_coefs, hid_coefs,
                                           nxt_coefs, rec_biases,
                                           out_coefs, out_biases, out_lp);
}